// DiceCEloss_43525198577795
// MI455X (gfx1250) — compile-verified
//
#include <hip/hip_runtime.h>
#include <hip/hip_bf16.h>

#define KCLS 21
#define HW (512 * 512)
#define NB 8
#define NBLOCKS1 256
#define CHUNKS_PER_B (NBLOCKS1 / NB)        // 32 blocks per batch image
#define PIX_PER_BLOCK (HW / CHUNKS_PER_B)   // 8192 pixels per block
#define ROWW 528                            // 33 * 16 columns (3*168 hists + 21 S + pad)
#define LOG2E 1.44269504088896340736f
#define LN2   0.69314718055994530942f

typedef __attribute__((ext_vector_type(2))) float v2f;
typedef __attribute__((ext_vector_type(8))) float v8f;

__device__ __forceinline__ float fexp2(float x) {
#if __has_builtin(__builtin_amdgcn_exp2f)
  return __builtin_amdgcn_exp2f(x);          // v_exp_f32
#else
  return exp2f(x);
#endif
}
__device__ __forceinline__ float flog2(float x) {
#if __has_builtin(__builtin_amdgcn_logf)
  return __builtin_amdgcn_logf(x);           // v_log_f32 (base-2)
#else
  return log2f(x);
#endif
}

// ---------------------------------------------------------------------------
// Kernel 1: streaming pass over pr/gt. Per-pixel argmax + logsumexp, LDS-atomic
// per-block histograms, then one dense zero-padded partial row per block.
// ---------------------------------------------------------------------------
__global__ __launch_bounds__(256) void dicece_pass1(
    const float* __restrict__ pr, const int* __restrict__ gt,
    float* __restrict__ ws)
{
  __shared__ unsigned sh_i[KCLS], sh_o[KCLS], sh_t[KCLS];
  __shared__ float sh_s[KCLS];
  const int tid = threadIdx.x;
  if (tid < KCLS) { sh_i[tid] = 0u; sh_o[tid] = 0u; sh_t[tid] = 0u; sh_s[tid] = 0.f; }
  __syncthreads();

  const int b     = blockIdx.x / CHUNKS_PER_B;
  const int chunk = blockIdx.x % CHUNKS_PER_B;
  const size_t prb = (size_t)b * KCLS * HW;
  const int* gtb   = gt + (size_t)b * HW;
  const int pix0   = chunk * PIX_PER_BLOCK;

  for (int it = 0; it < PIX_PER_BLOCK / (256 * 4); ++it) {
    const int pix = pix0 + ((it * 256 + tid) << 2);
    const int4 g4 = *reinterpret_cast<const int4*>(gtb + pix);

    float va[4][KCLS];                      // 4 pixels per lane, b128 loads
#pragma unroll
    for (int k = 0; k < KCLS; ++k) {
      const float4 x = *reinterpret_cast<const float4*>(pr + prb + (size_t)k * HW + pix);
      va[0][k] = x.x; va[1][k] = x.y; va[2][k] = x.z; va[3][k] = x.w;
    }
    const int gs[4] = {g4.x, g4.y, g4.z, g4.w};

#pragma unroll
    for (int s = 0; s < 4; ++s) {
      const int g = gs[s];
      float m = va[s][0]; int am = 0; float vg = va[s][0];
#pragma unroll
      for (int k = 1; k < KCLS; ++k) {
        const float x = va[s][k];
        const bool c = x > m;
        am = c ? k : am;
        m  = c ? x : m;
        vg = (k == g) ? x : vg;             // gather logit at gt class w/o dyn indexing
      }
      float sum = 0.f;
#pragma unroll
      for (int k = 0; k < KCLS; ++k)
        sum += fexp2((va[s][k] - m) * LOG2E);
      const float logp = vg - m - LN2 * flog2(sum);   // log_softmax at gt channel

      atomicAdd(&sh_t[g], 1u);              // ds_add_u32
      atomicAdd(&sh_o[am], 1u);
      if (am == g) atomicAdd(&sh_i[g], 1u);
      atomicAdd(&sh_s[g], logp);            // ds_add_f32
    }
  }
  __syncthreads();

  // Emit one dense 528-wide row: zeros everywhere except this block's slots.
  float* row = ws + (size_t)blockIdx.x * ROWW;
  for (int c = tid; c < ROWW; c += 256) {
    bool mine;
    if (c >= 3 * NB * KCLS) {
      mine = (c < 3 * NB * KCLS + KCLS);
    } else {
      const int off = c % (NB * KCLS);
      mine = (off >= b * KCLS) && (off < (b + 1) * KCLS);
    }
    if (!mine) row[c] = 0.f;
  }
  if (tid < KCLS) {
    row[                 b * KCLS + tid] = (float)sh_i[tid];
    row[    NB * KCLS +  b * KCLS + tid] = (float)sh_o[tid];
    row[2 * NB * KCLS +  b * KCLS + tid] = (float)sh_t[tid];
    row[3 * NB * KCLS + tid]             = sh_s[tid];
  }
}

// ---------------------------------------------------------------------------
// Kernel 2: column-sum the 256x528 partials with V_WMMA_F32_16X16X4_F32
// (A = all-ones => every row of C is the column sum; layout-permutation safe),
// then finalize dice weights + weighted CE to a scalar.
// ---------------------------------------------------------------------------
__global__ __launch_bounds__(256) void dicece_pass2(
    const float* __restrict__ ws, float* __restrict__ out)
{
  __shared__ float red[ROWW];
  __shared__ float shw[KCLS], shws[KCLS], shwc[KCLS];
  const int tid  = threadIdx.x;
  const int wv   = tid >> 5;
  const int lane = tid & 31;
  const int half = lane >> 4;
  const int nn   = lane & 15;
  const v2f a = {1.f, 1.f};                 // all-ones A (16x4)

  for (int t = wv; t < ROWW / 16; t += 8) { // 33 column tiles over 8 waves
    const int col = t * 16 + nn;
    v8f c = {0.f, 0.f, 0.f, 0.f, 0.f, 0.f, 0.f, 0.f};
    for (int r = 0; r < NBLOCKS1; r += 4) {
#if __has_builtin(__builtin_amdgcn_wmma_f32_16x16x4_f32)
      v2f bb;
      bb[0] = ws[(size_t)(r + 2 * half    ) * ROWW + col];
      bb[1] = ws[(size_t)(r + 2 * half + 1) * ROWW + col];
      c = __builtin_amdgcn_wmma_f32_16x16x4_f32(false, a, false, bb,
                                                (short)0, c, false, false);
#else
      c[0] += ws[(size_t)(r + 0) * ROWW + col] + ws[(size_t)(r + 1) * ROWW + col]
            + ws[(size_t)(r + 2) * ROWW + col] + ws[(size_t)(r + 3) * ROWW + col];
#endif
    }
    if (lane < 16) red[t * 16 + nn] = c[0]; // C row M=0 = column sums
  }
  __syncthreads();

  if (tid < KCLS) {
    float dice = 0.f, csum = 0.f;
    for (int b = 0; b < NB; ++b) {
      const float I  = red[                 b * KCLS + tid];
      const float AO = red[    NB * KCLS +  b * KCLS + tid];
      const float AT = red[2 * NB * KCLS +  b * KCLS + tid];
      dice += 2.f * I / (AO + AT + 1e-10f); // union + inter == AO + AT
      csum += AT;
    }
    dice *= (1.f / 8.f);                    // / SAMPLES
    const float w = 1.f - dice;
    shw[tid]  = w;
    shws[tid] = w * red[3 * NB * KCLS + tid];
    shwc[tid] = w * csum;
  }
  __syncthreads();

  if (tid == 0) {
    float sw = 0.f, ss = 0.f, sc = 0.f;
    for (int k = 0; k < KCLS; ++k) { sw += shw[k]; ss += shws[k]; sc += shwc[k]; }
    // BETA * mean(weight) + (-(sum w*logp) / (sum w*count))
    out[0] = sw * (1.f / KCLS) - ss / sc;
  }
}

extern "C" void kernel_launch(void* const* d_in, const int* in_sizes, int n_in,
                              void* d_out, int out_size, void* d_ws, size_t ws_size,
                              hipStream_t stream) {
  const float* pr = (const float*)d_in[0];
  const int*   gt = (const int*)d_in[1];
  float* ws = (float*)d_ws;                 // uses 256*528*4 = 540,672 bytes

  dicece_pass1<<<NBLOCKS1, 256, 0, stream>>>(pr, gt, ws);
  dicece_pass2<<<1, 256, 0, stream>>>(ws, (float*)d_out);

  (void)in_sizes; (void)n_in; (void)out_size; (void)ws_size;
}